// AcousticRadianceTransfer_PatchToPatch_89893665505845
// MI455X (gfx1250) — compile-verified
//
#include <hip/hip_runtime.h>
#include <hip/hip_bf16.h>
#include <math.h>
#include <stdint.h>

// ---------------- problem constants (from reference) ----------------
#define SRATE     16000
#define LOG_GAMMA (-6.9077552789821368f)   // ln(0.001)
#define NBOUNCE   8
#define NRAD      24000
#define NNZ_C     480000
#define LLEN      1600
#define NPATCH    256
#define NSLICE    60        // k-slices for WMMA reduction: 24000/60 = 400 (divisible by 4)
#define SLICE_K   (NRAD / NSLICE)

typedef float v2f __attribute__((ext_vector_type(2)));
typedef float v8f __attribute__((ext_vector_type(8)));

// ---------------- K0: zero counts + echo accumulator ----------------
__global__ void k_init(int* __restrict__ counts, float* __restrict__ echo) {
    int i = blockIdx.x * blockDim.x + threadIdx.x;
    int stride = gridDim.x * blockDim.x;
    for (int j = i; j < NRAD; j += stride) counts[j] = 0;
    for (int j = i; j < LLEN; j += stride) echo[j] = 0.f;
}

// ---------------- K1: row histogram ----------------
__global__ void k_hist(const int* __restrict__ row, int* __restrict__ counts) {
    int i = blockIdx.x * blockDim.x + threadIdx.x;
    int stride = gridDim.x * blockDim.x;
    for (int e = i; e < NNZ_C; e += stride) atomicAdd(&counts[row[e]], 1);
}

// ---------------- K2: exclusive scan (single block), writes row_start & cursor ----------------
__global__ void k_scan(const int* __restrict__ counts, int* __restrict__ row_start,
                       int* __restrict__ cursor) {
    __shared__ int sdata[1024];
    __shared__ int scarry;
    int tid = threadIdx.x;
    if (tid == 0) scarry = 0;
    __syncthreads();
    for (int base = 0; base < NRAD; base += 1024) {
        int i = base + tid;
        int v = (i < NRAD) ? counts[i] : 0;
        sdata[tid] = v;
        __syncthreads();
        for (int off = 1; off < 1024; off <<= 1) {
            int t = (tid >= off) ? sdata[tid - off] : 0;
            __syncthreads();
            sdata[tid] += t;
            __syncthreads();
        }
        int incl = sdata[tid];
        int excl = incl - v;
        int my_carry = scarry;
        if (i < NRAD) { row_start[i] = my_carry + excl; cursor[i] = my_carry + excl; }
        __syncthreads();
        if (tid == 1023) scarry += incl;
        __syncthreads();
    }
    if (tid == 0) row_start[NRAD] = scarry;
}

// ---------------- K3: scatter to CSR order, fold gain into weight ----------------
__global__ void k_scatter(const int* __restrict__ row, const int* __restrict__ col,
                          const int* __restrict__ refl, const float* __restrict__ basis,
                          const float* __restrict__ absorb, const float* __restrict__ scat,
                          const int* __restrict__ delay, int* __restrict__ cursor,
                          int* __restrict__ pcol, float* __restrict__ pw, int* __restrict__ pdel) {
    int i = blockIdx.x * blockDim.x + threadIdx.x;
    int stride = gridDim.x * blockDim.x;
    for (int e = i; e < NNZ_C; e += stride) {
        int r = row[e], c = col[e], rid = refl[e];
        float s0 = scat[rid], s1 = scat[NPATCH + rid];
        float v = (s0 * basis[e] + s1 * basis[NNZ_C + e]) * absorb[rid];
        int dd = delay[c];
        float wf = v * __expf(LOG_GAMMA * (float)dd / (float)SRATE);
        int pos = atomicAdd(&cursor[r], 1);
        pcol[pos] = c; pw[pos] = wf; pdel[pos] = dd;
    }
}

// ---------------- K4: X0 = initial_radiance * fsm ----------------
__global__ void k_x0(const float* __restrict__ init, float* __restrict__ cur) {
    long long i = (long long)blockIdx.x * blockDim.x + threadIdx.x;
    long long stride = (long long)gridDim.x * blockDim.x;
    const long long total = (long long)NRAD * LLEN;
    for (long long j = i; j < total; j += stride) {
        int t = (int)(j % LLEN);
        cur[j] = init[j] * __expf(LOG_GAMMA * (float)t / (float)SRATE);
    }
}

// -------- async copy of one 1600-float row: global -> LDS via ASYNCcnt path --------
// Low 32 bits of a flat pointer into __shared__ are the LDS byte offset
// (ISA 10.2: an LDS-aperture address keeps only its low 32 bits).
__device__ __forceinline__ void async_copy_row(const float* __restrict__ gsrc,
                                               float* lds_dst, int tid) {
    unsigned lds_base = (unsigned)(uintptr_t)lds_dst;
    #pragma unroll
    for (int j0 = 0; j0 < 400; j0 += 320) {        // 400 x b128 chunks
        int j = j0 + tid;
        if (j < 400) {
            unsigned laddr = lds_base + (unsigned)(j * 16);
            const float* g = gsrc + (size_t)j * 4;
            asm volatile("global_load_async_to_lds_b128 %0, %1, off"
                         :: "v"(laddr), "v"(g)
                         : "memory");
        }
    }
}

__device__ __forceinline__ void wait_async_all() {
    asm volatile("s_wait_asynccnt 0x0" ::: "memory");
}

// ---------------- K5: SpMM bounce, double-buffered async LDS staging ----------------
__global__ void __launch_bounds__(320)
k_spmm(const float* __restrict__ cur, float* __restrict__ nxt,
       const int* __restrict__ row_start, const int* __restrict__ pcol,
       const float* __restrict__ pw, const int* __restrict__ pdel) {
    __shared__ float lrow[2][LLEN];
    int r = blockIdx.x;
    int tid = threadIdx.x;
    int beg = row_start[r], end = row_start[r + 1];
    float acc[5] = {0.f, 0.f, 0.f, 0.f, 0.f};

    if (beg < end)
        async_copy_row(cur + (size_t)pcol[beg] * LLEN, lrow[0], tid);

    for (int i = beg; i < end; ++i) {
        int p = (i - beg) & 1;
        float w = pw[i];
        int dd = pdel[i];
        wait_async_all();          // my buffer-p fill complete
        __syncthreads();           // everyone's fill complete; prior readers of p^1 done
        if (i + 1 < end)
            async_copy_row(cur + (size_t)pcol[i + 1] * LLEN, lrow[p ^ 1], tid);
        #pragma unroll
        for (int j = 0; j < 5; ++j) {
            int t = tid + j * 320;
            int tt = t - dd;
            if (tt < 0) tt += LLEN;            // d in [0,1500), single wrap is enough
            acc[j] += w * lrow[p][tt];
        }
    }
    #pragma unroll
    for (int j = 0; j < 5; ++j)
        nxt[(size_t)r * LLEN + tid + j * 320] = acc[j];
}

// ---------------- K6: echo reduction via V_WMMA_F32_16X16X4_F32 ----------------
// part[s][t] = sum over k in slice s of detw[k] * X[k][t]  (row 0 of the 16x16 C tile).
// A fragment built branch-free: msk in {0,1} per lane, weights staged in LDS.
__global__ void __launch_bounds__(32)
k_reduce_wmma(const float* __restrict__ X, const float* __restrict__ dw,
              float* __restrict__ part) {
    __shared__ float sdw[SLICE_K];
    const int lane = threadIdx.x;           // wave32: one wave per block
    const int half = lane >> 4;             // 0: lanes 0-15, 1: lanes 16-31
    const int l15  = lane & 15;
    const int t0   = blockIdx.x * 16;       // 100 time tiles
    const int k0s  = blockIdx.y * SLICE_K;  // 400 rows per slice

    for (int j = lane; j < SLICE_K; j += 32) sdw[j] = dw[k0s + j];
    __syncthreads();

    // A (16x4 f32): row M=0 carries detw; other rows zero.
    // lanes 0-15 = M0-15 {K0,K1}; lanes 16-31 = M0-15 {K2,K3} -> only l15==0 is nonzero.
    const float msk = (l15 == 0) ? 1.f : 0.f;

    v8f c = {0.f, 0.f, 0.f, 0.f, 0.f, 0.f, 0.f, 0.f};

    #pragma unroll 4
    for (int kk = 0; kk < SLICE_K; kk += 4) {
        int ka = kk + (half ? 2 : 0);
        v2f a;
        a[0] = msk * sdw[ka];
        a[1] = msk * sdw[ka + 1];
        // B (4x16 f32): VGPR0: K0 (lanes 0-15) / K2 (lanes 16-31); VGPR1: K1 / K3.
        size_t kb = (size_t)(k0s + ka);
        v2f b;
        b[0] = X[kb * LLEN + t0 + l15];
        b[1] = X[(kb + 1) * LLEN + t0 + l15];
        c = __builtin_amdgcn_wmma_f32_16x16x4_f32(false, a, false, b, (short)0, c,
                                                  false, false);
    }
    // C/D layout: VGPR0 of lanes 0-15 holds M=0, N=lane -> the echo partials we want.
    if (lane < 16) part[(size_t)blockIdx.y * LLEN + t0 + lane] = c[0];
}

// ---------------- K7: deterministic slice combine ----------------
__global__ void k_part_sum(const float* __restrict__ part, float* __restrict__ echo) {
    int t = blockIdx.x * blockDim.x + threadIdx.x;
    if (t >= LLEN) return;
    float s = 0.f;
    for (int sl = 0; sl < NSLICE; ++sl) s += part[(size_t)sl * LLEN + t];
    echo[t] += s;
}

// ---------------- K8: final scaling ----------------
__global__ void k_final(const float* __restrict__ echo, const float* __restrict__ env,
                        float* __restrict__ out) {
    int t = blockIdx.x * blockDim.x + threadIdx.x;
    if (t >= LLEN) return;
    // /fsm[t] * exp(env[t]) == exp(env[t] - ln(gamma)*t/SR)
    out[t] = echo[t] * __expf(env[t] - LOG_GAMMA * (float)t / (float)SRATE);
}

// ---------------- host driver ----------------
extern "C" void kernel_launch(void* const* d_in, const int* in_sizes, int n_in,
                              void* d_out, int out_size, void* d_ws, size_t ws_size,
                              hipStream_t stream) {
    const float* absorb = (const float*)d_in[0];   // (256)
    const float* scat   = (const float*)d_in[1];   // (2,256)
    const float* init   = (const float*)d_in[2];   // (24000,1600)
    const float* detw   = (const float*)d_in[3];   // (24000)
    const float* env    = (const float*)d_in[4];   // (1600)
    const float* basis  = (const float*)d_in[5];   // (2,480000)
    const int*   row    = (const int*)d_in[6];     // (480000)
    const int*   col    = (const int*)d_in[7];     // (480000)
    const int*   refl   = (const int*)d_in[8];     // (480000)
    const int*   delay  = (const int*)d_in[9];     // (24000)
    float* out = (float*)d_out;                    // (1600)

    // workspace carve-up (~313 MB total; f32 time-domain state stays L2-resident)
    float* cur       = (float*)d_ws;
    float* nxt       = cur + (size_t)NRAD * LLEN;
    int*   pcol      = (int*)(nxt + (size_t)NRAD * LLEN);
    float* pw        = (float*)(pcol + NNZ_C);
    int*   pdel      = (int*)(pw + NNZ_C);
    int*   row_start = pdel + NNZ_C;
    int*   cursor    = row_start + (NRAD + 1);
    float* echo      = (float*)(cursor + NRAD);
    float* part      = echo + LLEN;                // NSLICE * LLEN floats

    // CSR build + init
    k_init<<<128, 256, 0, stream>>>(cursor, echo);                       // cursor reused as counts
    k_hist<<<512, 256, 0, stream>>>(row, cursor);
    k_scan<<<1, 1024, 0, stream>>>(cursor, row_start, cursor);
    k_scatter<<<512, 256, 0, stream>>>(row, col, refl, basis, absorb, scat, delay,
                                       cursor, pcol, pw, pdel);

    // X0 = init * fsm
    k_x0<<<1024, 256, 0, stream>>>(init, cur);

    dim3 rgrid(LLEN / 16, NSLICE);
    // term 0 (R0 contribution)
    k_reduce_wmma<<<rgrid, 32, 0, stream>>>(cur, detw, part);
    k_part_sum<<<(LLEN + 255) / 256, 256, 0, stream>>>(part, echo);

    float* a = cur;
    float* b = nxt;
    for (int bnc = 0; bnc < NBOUNCE; ++bnc) {
        k_spmm<<<NRAD, 320, 0, stream>>>(a, b, row_start, pcol, pw, pdel);
        float* tmp = a; a = b; b = tmp;
        k_reduce_wmma<<<rgrid, 32, 0, stream>>>(a, detw, part);
        k_part_sum<<<(LLEN + 255) / 256, 256, 0, stream>>>(part, echo);
    }

    k_final<<<(LLEN + 255) / 256, 256, 0, stream>>>(echo, env, out);
}